// PostProcessor_26998164423214
// MI455X (gfx1250) — compile-verified
//
#include <hip/hip_runtime.h>
#include <hip/hip_bf16.h>
#include <stdint.h>

#define NUM_Q   300
#define NUM_C   80
#define QC      (NUM_Q * NUM_C)      // 24000 scores per batch row
#define TOPK    300
#define BLOCK   256
#define SORTN   512                  // next pow2 >= TOPK

// order-preserving float->uint mapping (larger float -> larger key)
__device__ __forceinline__ uint32_t fkey(float f) {
    uint32_t u = __float_as_uint(f);
    return (u & 0x80000000u) ? ~u : (u | 0x80000000u);
}
__device__ __forceinline__ float keyf(uint32_t k) {
    uint32_t u = (k & 0x80000000u) ? (k & 0x7FFFFFFFu) : ~k;
    return __uint_as_float(u);
}

__global__ __launch_bounds__(BLOCK)
void postprocess_kernel(const float* __restrict__ pred_logits,
                        const float* __restrict__ pred_boxes,
                        const float* __restrict__ orig_sizes,
                        float* __restrict__ out, int B) {
    __shared__ uint32_t sKeys[QC];       // 96000 B: raw logits then score-keys
    __shared__ uint64_t sCand[SORTN];    // 4096 B : top-K composite keys
    __shared__ uint32_t sHist[256];      // 1024 B : radix histogram
    __shared__ uint64_t sPref;
    __shared__ uint32_t sRem;
    __shared__ uint32_t sCnt;

    const uint32_t tid = threadIdx.x;
    const int b = blockIdx.x;
    if (b >= B) return;

    // ---- 1) async global->LDS copy of this row's 24000 logits (6000 x 16B) ----
    {
        uint64_t gbase = (uint64_t)(uintptr_t)(pred_logits + (size_t)b * QC);
        uint32_t lbase = (uint32_t)(uintptr_t)(void*)sKeys;  // low 32 bits of flat shared ptr = LDS byte addr
        for (uint32_t c = tid; c < (QC * 4u) / 16u; c += BLOCK) {
            uint32_t laddr = lbase + c * 16u;
            uint64_t gaddr = gbase + (uint64_t)c * 16u;
            asm volatile("global_load_async_to_lds_b128 %0, %1, off"
                         :: "v"(laddr), "v"(gaddr) : "memory");
        }
        asm volatile("s_wait_asynccnt 0" ::: "memory");
    }
    __syncthreads();

    // ---- 2) sigmoid + order key, in place ----
    for (uint32_t i = tid; i < QC; i += BLOCK) {
        float x = __uint_as_float(sKeys[i]);
        float s = 1.0f / (1.0f + expf(-x));
        sKeys[i] = fkey(s);
    }
    if (tid == 0) { sPref = 0ull; sRem = TOPK; sCnt = 0u; }
    __syncthreads();

    // ---- 3) exact radix-select (MSB-first, 8 passes x 8 bits) on
    //         key64 = (scoreKey<<32) | (0xFFFFFFFF - index)  (all unique) ----
    for (int p = 0; p < 8; ++p) {
        sHist[tid & 255u] = 0u;
        __syncthreads();
        const uint64_t pref = sPref;
        const int knownShift = 64 - 8 * p;   // guarded: only used when p > 0
        const int digShift   = 56 - 8 * p;
        for (uint32_t i = tid; i < QC; i += BLOCK) {
            uint64_t k64 = ((uint64_t)sKeys[i] << 32) | (uint64_t)(0xFFFFFFFFu - i);
            bool match = (p == 0) || ((k64 >> knownShift) == pref);
            if (match) {
                uint32_t d = (uint32_t)((k64 >> digShift) & 0xFFull);
                atomicAdd(&sHist[d], 1u);
            }
        }
        __syncthreads();
        if (tid == 0) {
            uint32_t rem = sRem, cum = 0; int d = 255;
            for (; d >= 0; --d) { cum += sHist[d]; if (cum >= rem) break; }
            if (d < 0) d = 0;                      // defensive; cannot happen
            sRem  = rem - (cum - sHist[d]);
            sPref = (pref << 8) | (uint64_t)(uint32_t)d;
        }
        __syncthreads();
    }
    const uint64_t T64 = sPref;   // exact 300th-largest composite key

    // ---- 4) compact the exactly-300 winners, pad, bitonic sort descending ----
    for (uint32_t i = tid; i < QC; i += BLOCK) {
        uint64_t k64 = ((uint64_t)sKeys[i] << 32) | (uint64_t)(0xFFFFFFFFu - i);
        if (k64 >= T64) {
            uint32_t pos = atomicAdd(&sCnt, 1u);
            if (pos < SORTN) sCand[pos] = k64;
        }
    }
    __syncthreads();
    for (uint32_t i = TOPK + tid; i < SORTN; i += BLOCK) sCand[i] = 0ull;

    for (uint32_t k = 2; k <= SORTN; k <<= 1) {
        for (uint32_t j = k >> 1; j > 0; j >>= 1) {
            __syncthreads();
            for (uint32_t i = tid; i < SORTN; i += BLOCK) {
                uint32_t l = i ^ j;
                if (l > i) {
                    uint64_t a = sCand[i], c = sCand[l];
                    bool up = ((i & k) == 0);          // descending overall
                    if (up ? (a < c) : (a > c)) { sCand[i] = c; sCand[l] = a; }
                }
            }
        }
    }
    __syncthreads();

    // ---- 5) emit boxes / labels / scores ----
    const float* brow = pred_boxes + (size_t)b * NUM_Q * 4;
    const float o0 = orig_sizes[(size_t)b * 2 + 0];
    const float o1 = orig_sizes[(size_t)b * 2 + 1];
    float* outBoxes  = out;
    float* outLabels = out + (size_t)B * TOPK * 4;
    float* outScores = outLabels + (size_t)B * TOPK;

    for (uint32_t j = tid; j < TOPK; j += BLOCK) {
        uint64_t k64 = sCand[j];
        uint32_t skey = (uint32_t)(k64 >> 32);
        uint32_t idx  = 0xFFFFFFFFu - (uint32_t)(k64 & 0xFFFFFFFFull);
        float score = keyf(skey);
        uint32_t q   = idx / NUM_C;
        uint32_t lab = idx - q * NUM_C;

        float cx = brow[q * 4 + 0], cy = brow[q * 4 + 1];
        float w  = brow[q * 4 + 2], h  = brow[q * 4 + 3];

        size_t ob = ((size_t)b * TOPK + j) * 4;
        outBoxes[ob + 0] = (cx - 0.5f * w) * o0;
        outBoxes[ob + 1] = (cy - 0.5f * h) * o1;
        outBoxes[ob + 2] = (cx + 0.5f * w) * o0;
        outBoxes[ob + 3] = (cy + 0.5f * h) * o1;
        outLabels[(size_t)b * TOPK + j] = (float)lab;
        outScores[(size_t)b * TOPK + j] = score;
    }
}

extern "C" void kernel_launch(void* const* d_in, const int* in_sizes, int n_in,
                              void* d_out, int out_size, void* d_ws, size_t ws_size,
                              hipStream_t stream) {
    (void)n_in; (void)out_size; (void)d_ws; (void)ws_size;
    const float* pred_logits = (const float*)d_in[0];
    const float* pred_boxes  = (const float*)d_in[1];
    const float* orig_sizes  = (const float*)d_in[2];
    float* out = (float*)d_out;
    int B = in_sizes[0] / QC;        // 2048 for the reference shapes
    dim3 grid((unsigned)B), block(BLOCK);
    postprocess_kernel<<<grid, block, 0, stream>>>(pred_logits, pred_boxes,
                                                   orig_sizes, out, B);
}